// CNPsLoss_14637248544954
// MI455X (gfx1250) — compile-verified
//
#include <hip/hip_runtime.h>

typedef __attribute__((ext_vector_type(2))) float v2f;
typedef __attribute__((ext_vector_type(4))) float v4f;
typedef __attribute__((ext_vector_type(8))) float v8f;

#define LOG2PI   1.8378770664093453f
#define LN2      0.6931471805599453f
#define INV_LN2  1.4426950408889634f

// t = log(softplus(ls)) + (tv-mu)^2 / softplus(ls)
// Uses raw v_exp_f32 / v_log_f32 (base-2) and v_rcp_f32 — no ocml call chains,
// no IEEE divide fixup sequence.
static __device__ __forceinline__ float t_elem(float mu, float ls, float tv) {
    float e    = __builtin_amdgcn_exp2f(-__builtin_fabsf(ls) * INV_LN2);
    float sp   = fmaxf(ls, 0.0f) + LN2 * __builtin_amdgcn_logf(1.0f + e);
    float diff = tv - mu;
    return LN2 * __builtin_amdgcn_logf(sp) + diff * diff * __builtin_amdgcn_rcpf(sp);
}

// Pass 1: each block produces one partial sum of t over its slice.
// Wave reduction runs on the matrix pipe: C += A(partials 16x4) x B(ones 4x16).
__global__ __launch_bounds__(256) void cnps_partial_kernel(
    const float* __restrict__ outputs,   // [B, 2D] row-major
    const float* __restrict__ targets,   // [B, 2D] row-major (only first D cols used)
    float* __restrict__ partial,         // [gridDim.x]
    int iters, int dshift, long ntot)
{
    const int  d_half = 1 << dshift;                    // D (power of two)
    const int  dmask  = d_half - 1;
    const int  tid    = blockIdx.x * blockDim.x + threadIdx.x;
    const long stride = (long)gridDim.x * blockDim.x;   // total threads

    v8f c = {0.f, 0.f, 0.f, 0.f, 0.f, 0.f, 0.f, 0.f};
    v2f ones; ones.x = 1.0f; ones.y = 1.0f;             // full 4x16 B matrix of ones

    // Uniform trip count across the whole block -> EXEC all-ones at every WMMA.
    #pragma unroll 2
    for (int it = 0; it < iters; ++it) {
        long i = ((long)it * stride + tid) * 4;         // first of 4 consecutive elems
        long b = i >> dshift;
        long d = i & dmask;
        long rowbase = (b << (dshift + 1)) + d;         // b*2D + d
        const v4f mu = __builtin_nontemporal_load((const v4f*)(outputs + rowbase));
        const v4f ls = __builtin_nontemporal_load((const v4f*)(outputs + rowbase + d_half));
        const v4f tv = __builtin_nontemporal_load((const v4f*)(targets + rowbase));

        float t0 = t_elem(mu.x, ls.x, tv.x);
        float t1 = t_elem(mu.y, ls.y, tv.y);
        float t2 = t_elem(mu.z, ls.z, tv.z);
        float t3 = t_elem(mu.w, ls.w, tv.w);

        v2f a; a.x = t0 + t1; a.y = t2 + t3;            // this lane's two A-matrix slots
        // v_wmma_f32_16x16x4_f32: D = A x ones + C  => C rows accumulate row-sums
        c = __builtin_amdgcn_wmma_f32_16x16x4_f32(
                /*neg_a=*/false, a, /*neg_b=*/false, ones,
                /*c_mod=*/(short)0, c, /*reuse_a=*/false, /*reuse_b=*/false);
    }

    // C rows are replicated across columns: summing the 8 C regs gives
    // sum(rows 0..7) in lanes [0,16) and sum(rows 8..15) in lanes [16,32).
    float s = c[0] + c[1] + c[2] + c[3] + c[4] + c[5] + c[6] + c[7];
    float wave_sum = s + __shfl_xor(s, 16, 32);         // full 32-lane (64-value) sum

    // Scalar tail for sizes that don't divide evenly (zero-trip for the given shape).
    float acc_t = 0.0f;
    long done = (long)iters * stride * 4;
    for (long i = done + tid; i < ntot; i += stride) {
        long b = i >> dshift;
        long d = i & dmask;
        long rowbase = (b << (dshift + 1)) + d;
        acc_t += t_elem(outputs[rowbase], outputs[rowbase + d_half], targets[rowbase]);
    }
    for (int o = 16; o > 0; o >>= 1) acc_t += __shfl_xor(acc_t, o, 32);
    wave_sum += acc_t;

    __shared__ float lsum[8];
    const int lane = threadIdx.x & 31;
    const int wv   = threadIdx.x >> 5;
    if (lane == 0) lsum[wv] = wave_sum;
    __syncthreads();
    if (threadIdx.x == 0) {
        float blk = 0.0f;
        #pragma unroll
        for (int k = 0; k < 8; ++k) blk += lsum[k];
        partial[blockIdx.x] = blk;
    }
}

// Pass 2: deterministic single-block tree reduction of block partials -> scalar loss.
__global__ __launch_bounds__(256) void cnps_final_kernel(
    const float* __restrict__ partial, int nparts,
    float* __restrict__ out, float inv_b, float add_const)
{
    __shared__ float sm[256];
    float s = 0.0f;
    for (int i = threadIdx.x; i < nparts; i += 256) s += partial[i];
    sm[threadIdx.x] = s;
    __syncthreads();
    for (int o = 128; o > 0; o >>= 1) {
        if (threadIdx.x < o) sm[threadIdx.x] += sm[threadIdx.x + o];
        __syncthreads();
    }
    if (threadIdx.x == 0)
        out[0] = 0.5f * sm[0] * inv_b + add_const;      // + 0.5*D*log(2pi)
}

extern "C" void kernel_launch(void* const* d_in, const int* in_sizes, int n_in,
                              void* d_out, int out_size, void* d_ws, size_t ws_size,
                              hipStream_t stream) {
    const float* outputs = (const float*)d_in[0];
    const float* targets = (const float*)d_in[1];
    float*       out     = (float*)d_out;
    float*       partial = (float*)d_ws;

    const int  two_d  = 1024;                           // fixed by reference
    const int  d_half = two_d / 2;
    const long total  = (long)in_sizes[0];              // B * 2D
    const int  Brows  = (int)(total / two_d);           // 16384
    const long ntot   = (long)Brows * d_half;           // B * D elements

    int dshift = 0;
    while ((1 << (dshift + 1)) <= d_half) ++dshift;     // log2(D) = 9

    const int  GRID = 1024, BLOCK = 256;
    const long stride = (long)GRID * BLOCK;
    const int  iters  = (int)(ntot / (stride * 4));     // 8 for the given shape

    cnps_partial_kernel<<<GRID, BLOCK, 0, stream>>>(outputs, targets, partial,
                                                    iters, dshift, ntot);

    const float inv_b     = 1.0f / (float)Brows;
    const float add_const = 0.5f * (float)d_half * LOG2PI;
    cnps_final_kernel<<<1, 256, 0, stream>>>(partial, GRID, out, inv_b, add_const);
}